// HouseholderLayer_25314537242638
// MI455X (gfx1250) — compile-verified
//
#include <hip/hip_runtime.h>
#include <stdint.h>

typedef __attribute__((ext_vector_type(2))) float v2f;
typedef __attribute__((ext_vector_type(4))) float v4f;
typedef __attribute__((ext_vector_type(8))) float v8f;
typedef __attribute__((ext_vector_type(4))) unsigned int v4u;
typedef __attribute__((ext_vector_type(4))) int v4i;
typedef __attribute__((ext_vector_type(8))) int v8i;

#define ROWS      16                 // rows per workgroup (one WMMA M-tile)
#define CHUNK     256                // K columns staged per TDM transfer
#define PAD_DW    10                 // LDS pad DWORDs per row (bank-conflict-free col gathers)
#define LDSTRIDE  (CHUNK + PAD_DW)   // 266 floats per LDS row
#define NTHREADS  256                // 8 wave32 per workgroup
#define NWAVES    (NTHREADS / 32)
#define EPS       1e-16f

// ---------------------------------------------------------------------------
// Kernel 1: s = sum(v^2)+eps ;  ws[0] = 2/s, ws[1] = 2*eps/s
// ---------------------------------------------------------------------------
__global__ __launch_bounds__(256)
void hh_scalars(const float* __restrict__ v, float* __restrict__ ws, int n) {
  __shared__ float red[256];
  const int tid = threadIdx.x;
  float s = 0.f;
  for (int i = tid; i < n; i += 256) { float t = v[i]; s += t * t; }
  red[tid] = s;
  __syncthreads();
  for (int off = 128; off > 0; off >>= 1) {
    if (tid < off) red[tid] += red[tid + off];
    __syncthreads();
  }
  if (tid == 0) {
    float ss = red[0] + EPS;
    ws[0] = 2.0f / ss;
    ws[1] = (2.0f * EPS) / ss;
  }
}

// ---------------------------------------------------------------------------
// TDM: DMA a ROWS x CHUNK f32 tile (row stride = n elems) into LDS,
// padding each 256-DWORD row with 10 DWORDs (LDS row stride = 266 floats).
// D# layout per CDNA5 ISA ch.8 (group0 128b, group1 256b; groups 2/3 unused).
// This toolchain's builtin takes 6 args (g0, g1, g2, g3, extra group, cpol).
// ---------------------------------------------------------------------------
__device__ __forceinline__ void tdm_load_tile(const float* gsrc, unsigned lds_off, int n) {
  const unsigned long long ga = (unsigned long long)(uintptr_t)gsrc;

  v4u g0;
  g0[0] = 1u;                                          // count=1 (valid), user mode
  g0[1] = lds_off;                                     // lds_addr (bytes)
  g0[2] = (unsigned)(ga & 0xffffffffull);              // global_addr[31:0]
  g0[3] = (unsigned)((ga >> 32) & 0x01ffffffull)       // global_addr[56:32]
        | (2u << 30);                                  // type = 2 ("image")

  const unsigned td0 = (unsigned)n;                    // tensor dim0 (row length)
  const unsigned td1 = (unsigned)ROWS;                 // tensor dim1 (tile rows)
  const unsigned long long st0 = (unsigned long long)(unsigned)n; // dim0 stride (elems)

  v8i g1;
  g1[0] = (int)((2u << 16)                             // data_size = 4 bytes
              | (1u << 20)                             // pad_enable
              | (7u << 22)                             // pad_interval: 256 DWORDs
              | ((unsigned)(PAD_DW - 1) << 25));       // pad_amount: 10 DWORDs
  g1[1] = (int)((td0 & 0xffffu) << 16);                // tensor_dim0[15:0]
  g1[2] = (int)((td0 >> 16) | ((td1 & 0xffffu) << 16));// tensor_dim0[31:16] | tensor_dim1[15:0]
  g1[3] = (int)((td1 >> 16) | ((unsigned)CHUNK << 16));// tensor_dim1[31:16] | tile_dim0
  g1[4] = (int)(unsigned)ROWS;                         // tile_dim1 | tile_dim2=0
  g1[5] = (int)(unsigned)(st0 & 0xffffffffull);        // tensor_dim0_stride[31:0]
  g1[6] = (int)(unsigned)((st0 >> 32) & 0xffffu);      // stride[47:32] | dim1_stride[15:0]=0
  g1[7] = 0;

  const v4i gz4 = {0, 0, 0, 0};
  const v8i gz8 = {0, 0, 0, 0, 0, 0, 0, 0};
  __builtin_amdgcn_tensor_load_to_lds(g0, g1, gz4, gz4, gz8, 0);
}

// ---------------------------------------------------------------------------
// Kernel 2: fused Householder apply for 16 rows per workgroup.
// Phase 1: TDM-staged tiles -> V_WMMA_F32_16X16X4_F32 computes, per row,
//          D[:,0] = x.v (B col0 = v chunk) and D[:,1] = rowsum (B col1 = 1).
// Phase 2: out = x - (2/s)(x.v) v - (2eps/s) rowsum + bias  (x re-read, L2-hot)
// ---------------------------------------------------------------------------
__global__ __launch_bounds__(NTHREADS)
void hh_main(const float* __restrict__ x, const float* __restrict__ v,
             const float* __restrict__ bias, const float* __restrict__ ws,
             float* __restrict__ out, int n) {
  __shared__ float tile[2][ROWS * LDSTRIDE];
  __shared__ float red[2][ROWS][NWAVES];
  __shared__ float coef[2][ROWS];

  const int tid    = threadIdx.x;
  const int lane   = tid & 31;
  const int wave   = tid >> 5;
  const int laneLo = lane & 15;
  const int laneHi = lane >> 4;
  const int nchunk = n / CHUNK;

  const long row0   = (long)blockIdx.x * ROWS;
  const float* xrow = x + row0 * (long)n;

  v8f acc = {0.f, 0.f, 0.f, 0.f, 0.f, 0.f, 0.f, 0.f};

  // prologue: wave 0 kicks off chunk 0
  if (tid < 32) {
    tdm_load_tile(xrow, (unsigned)(uintptr_t)&tile[0][0], n);
  }

  for (int c = 0; c < nchunk; ++c) {
    if (tid < 32) {
      if (c + 1 < nchunk) {
        // double-buffer: issue next chunk, then wait for current one
        tdm_load_tile(xrow + (long)(c + 1) * CHUNK,
                      (unsigned)(uintptr_t)&tile[(c + 1) & 1][0], n);
        __builtin_amdgcn_s_wait_tensorcnt(1);
      } else {
        __builtin_amdgcn_s_wait_tensorcnt(0);
      }
    }
    __syncthreads();  // chunk c visible to all waves

    const float* tl      = &tile[c & 1][0];
    const int    colbase = wave * (CHUNK / NWAVES) + 2 * laneHi;
    const float* vch     = v + (long)c * CHUNK;

#pragma unroll
    for (int kk = 0; kk < CHUNK / NWAVES / 4; ++kk) {
      const int col = colbase + kk * 4;
      // A frag (16x4 f32): lanes 0-15 K=k,k+1 ; lanes 16-31 K=k+2,k+3
      v2f a = *(const v2f*)&tl[laneLo * LDSTRIDE + col];
      // B frag (4x16 f32): N=0 column = v, N=1 column = ones, rest zero
      v2f vv = *(const v2f*)&vch[col];
      v2f bfrag;
      bfrag.x = (laneLo == 0) ? vv.x : ((laneLo == 1) ? 1.0f : 0.0f);
      bfrag.y = (laneLo == 0) ? vv.y : ((laneLo == 1) ? 1.0f : 0.0f);
      acc = __builtin_amdgcn_wmma_f32_16x16x4_f32(false, a, false, bfrag,
                                                  (short)0, acc, false, false);
    }
    __syncthreads();  // all waves done reading buf before TDM overwrites it
  }

  // D layout: lane laneLo = N column; VGPR r = row r (laneHi=0) / row 8+r (laneHi=1)
  if (laneLo < 2) {
#pragma unroll
    for (int r = 0; r < 8; ++r)
      red[laneLo][laneHi * 8 + r][wave] = acc[r];
  }
  __syncthreads();

  if (tid < ROWS) {
    float xv = 0.f, rs = 0.f;
#pragma unroll
    for (int w = 0; w < NWAVES; ++w) { xv += red[0][tid][w]; rs += red[1][tid][w]; }
    coef[0][tid] = ws[0] * xv;   // (2/s) * (x.v)
    coef[1][tid] = ws[1] * rs;   // (2eps/s) * rowsum
  }
  __syncthreads();

  // Phase 2: elementwise update, float4-vectorized; x re-read hits L2.
  const int  q   = n >> 2;
  const v4f* x4  = (const v4f*)xrow;
  v4f*       o4  = (v4f*)(out + row0 * (long)n);
  const v4f* v4p = (const v4f*)v;
  const v4f* b4p = (const v4f*)bias;
  const int total = ROWS * q;
  for (int i = tid; i < total; i += NTHREADS) {
    const int r    = i / q;
    const int cidx = i - r * q;
    v4f xx = x4[i];
    v4f vv = v4p[cidx];
    v4f bb = b4p[cidx];
    const float a  = coef[0][r];
    const float cc = coef[1][r];
    v4f oo = xx - a * vv - cc + bb;
    o4[i] = oo;
  }
}

// ---------------------------------------------------------------------------
extern "C" void kernel_launch(void* const* d_in, const int* in_sizes, int n_in,
                              void* d_out, int out_size, void* d_ws, size_t ws_size,
                              hipStream_t stream) {
  const float* x    = (const float*)d_in[0];
  const float* v    = (const float*)d_in[1];
  const float* bias = (const float*)d_in[2];
  float* out = (float*)d_out;
  float* ws  = (float*)d_ws;

  const int  n     = in_sizes[1];                 // N = 4096
  const long brows = (long)in_sizes[0] / n;       // B = 16384

  hh_scalars<<<1, 256, 0, stream>>>(v, ws, n);
  hh_main<<<(int)(brows / ROWS), NTHREADS, 0, stream>>>(x, v, bias, ws, out, n);
}